// ModelNew_50388556316978
// MI455X (gfx1250) — compile-verified
//
#include <hip/hip_runtime.h>

typedef __attribute__((ext_vector_type(2))) float v2f;
typedef __attribute__((ext_vector_type(8))) float v8f;

#define CIN   32
#define HIN   128
#define WIN   256
#define GRP   4
#define ICG   8
#define OCG   16
#define KH_   3
#define KW_   5
#define HOUT  257
#define WOUT  766
#define HW    (HIN * WIN)
#define WPG   (ICG * OCG * KH_ * KW_)   /* 1920 weights per group */

// Zero the 129 even output rows (oh = 2*ih - 1 + 2*kh is always odd).
__global__ __launch_bounds__(256) void zero_even_rows_kernel(float* __restrict__ out) {
    const int row = blockIdx.x;            // 1024 channel-images * 129 even rows
    const int bc  = row / 129;
    const int ohe = row - bc * 129;
    float* p = out + ((size_t)bc * HOUT + 2 * ohe) * WOUT;
    for (int i = threadIdx.x; i < WOUT; i += blockDim.x) p[i] = 0.0f;
}

// Implicit-GEMM transposed conv on odd output rows via V_WMMA_F32_16X16X4_F32.
// One wave = one (b, g, odd-row, 48-column tile); 3 accumulators (phase r=ow%3).
// All 8 waves of a block share the same group g -> group weights staged in LDS,
// repacked [ic*16+oc][kh*8+kw] (kw padded to 8) for aligned ds_load_b128.
__global__ __launch_bounds__(256) void convtr_wmma_kernel(const float* __restrict__ x,
                                                          const float* __restrict__ w,
                                                          float* __restrict__ out) {
    __shared__ float lw[ICG * OCG * KH_ * 8];   // 128 rows * 24 floats = 12.3 KB

    const int g_blk = (blockIdx.x >> 8) & 3;    // uniform across block
    {
        const float* wgrp = w + (size_t)g_blk * WPG;
        for (int i = threadIdx.x; i < WPG; i += 256) {
            const int icoc = i / (KH_ * KW_);
            const int rem  = i - icoc * (KH_ * KW_);
            lw[icoc * 24 + (rem / KW_) * 8 + (rem % KW_)] = wgrp[i];
        }
    }
    __syncthreads();

    const int lane  = threadIdx.x & 31;
    const int wave  = threadIdx.x >> 5;
    const int idx   = blockIdx.x * 8 + wave;     // 2^17 waves total
    const int t     = idx & 15;                  // 48-col tile (ow base = 48*t)
    const int oh_i  = (idx >> 4) & 127;          // odd row: oh = 2*oh_i + 1
    const int g     = (idx >> 11) & 3;           // == g_blk
    const int b     = idx >> 13;
    const int oh    = 2 * oh_i + 1;

    const int ncol  = lane & 15;                 // N (and A's M) index
    const int khalf = lane >> 4;                 // 0: K={0,1}, 1: K={2,3}
    const int k_in  = khalf << 1;

    v8f acc0 = {}; v8f acc1 = {}; v8f acc2 = {};

    const float* xg = x + (size_t)(b * CIN + g * ICG) * HW;

    // Per-lane B column: q0 index iw = 16t + ncol, q1 index iw+1.
    // Only iw==255 can make iw+1 out of range -> clamp + select (no exec branch).
    const int  iw    = 16 * t + ncol;
    const int  iwb   = (iw > WIN - 2) ? (WIN - 2) : iw;   // min(iw, 254)
    const bool edge  = (iw == WIN - 1);

#pragma unroll
    for (int kh = 0; kh < KH_; ++kh) {
        const int ih = oh_i + 1 - kh;                    // oh = 2*ih - 1 + 2*kh
        if ((unsigned)ih >= (unsigned)HIN) continue;     // wave-uniform
        const float* xrow = xg + (size_t)ih * WIN;
#pragma unroll
        for (int c = 0; c < 2; ++c) {
            const int ic0 = c * 4 + k_in;                // K -> input channel

            // B fragments: two shifts (q=0,1) serve all five (kw,r) taps.
            const float* p0 = xrow + (size_t)ic0 * HW + iwb;
            const float alo = p0[0],  ahi = p0[1];       // channel ic0
            const float blo = p0[HW], bhi = p0[HW + 1];  // channel ic0+1
            v2f bq0, bq1;
            bq0.x = edge ? ahi : alo;   bq1.x = edge ? 0.0f : ahi;
            bq0.y = edge ? bhi : blo;   bq1.y = edge ? 0.0f : bhi;

            // A fragments from LDS: kw 0..3 as b128, kw4 as b32, per channel.
            const float* l0 = &lw[(ic0 * OCG + ncol) * 24 + kh * 8];
            const float* l1 = l0 + OCG * 24;
            const float4 wa = *reinterpret_cast<const float4*>(l0);
            const float  wa4 = l0[4];
            const float4 wb = *reinterpret_cast<const float4*>(l1);
            const float  wb4 = l1[4];

            // taps: r=0:(kw2,q0)  r=1:(kw0,q1),(kw3,q0)  r=2:(kw1,q1),(kw4,q0)
            v2f a;
            a.x = wa.z; a.y = wb.z;
            acc0 = __builtin_amdgcn_wmma_f32_16x16x4_f32(false, a, false, bq0,
                                                         (short)0, acc0, false, false);
            a.x = wa.x; a.y = wb.x;
            acc1 = __builtin_amdgcn_wmma_f32_16x16x4_f32(false, a, false, bq1,
                                                         (short)0, acc1, false, false);
            a.x = wa.w; a.y = wb.w;
            acc1 = __builtin_amdgcn_wmma_f32_16x16x4_f32(false, a, false, bq0,
                                                         (short)0, acc1, false, false);
            a.x = wa.y; a.y = wb.y;
            acc2 = __builtin_amdgcn_wmma_f32_16x16x4_f32(false, a, false, bq1,
                                                         (short)0, acc2, false, false);
            a.x = wa4; a.y = wb4;
            acc2 = __builtin_amdgcn_wmma_f32_16x16x4_f32(false, a, false, bq0,
                                                         (short)0, acc2, false, false);
        }
    }

    // Store: lane ncol covers ow0 = 48*t + 3*ncol, phases r=0,1,2 contiguous;
    // 16 lanes -> 192 consecutive bytes per channel-row (fully coalesced b96).
    struct __attribute__((packed, aligned(4))) f3 { float x, y, z; };
    const int  mbase = khalf * 8;
    const int  ow0   = 48 * t + 3 * ncol;
    float* og = out + ((size_t)(b * (GRP * OCG) + g * OCG) * HOUT + oh) * WOUT;
#pragma unroll
    for (int v = 0; v < 8; ++v) {
        const int oc = mbase + v;
        float* op = og + (size_t)oc * (HOUT * WOUT) + ow0;
        if (ow0 + 2 < WOUT) {
            f3 val = { acc0[v], acc1[v], acc2[v] };
            *reinterpret_cast<f3*>(op) = val;
        } else {                               // only tile t=15, lane 15
            if (ow0     < WOUT) op[0] = acc0[v];
            if (ow0 + 1 < WOUT) op[1] = acc1[v];
            if (ow0 + 2 < WOUT) op[2] = acc2[v];
        }
    }
}

extern "C" void kernel_launch(void* const* d_in, const int* in_sizes, int n_in,
                              void* d_out, int out_size, void* d_ws, size_t ws_size,
                              hipStream_t stream) {
    const float* x = (const float*)d_in[0];   // (16,32,128,256)
    const float* w = (const float*)d_in[1];   // (32,16,3,5)
    float* out = (float*)d_out;               // (16,64,257,766)

    zero_even_rows_kernel<<<1024 * 129, 256, 0, stream>>>(out);
    convtr_wmma_kernel<<<16384, 256, 0, stream>>>(x, w, out);
}